// WENO_Z_23802708754741
// MI455X (gfx1250) — compile-verified
//
#include <hip/hip_runtime.h>
#include <stdint.h>

// WENO-Z on MI455X (gfx1250, wave32):
//  - linear stencil part (9 combos x 5 taps) via V_WMMA_F32_16X16X4_F32
//  - x staged into LDS per block via global_load_async_to_lds_b32 (ASYNCcnt)
//  - nonlinear WENO weights on VALU, one lane per output element

#define NB 16
#define NN (1 << 20)
#define NMASK (NN - 1)
#define BLOCK_THREADS 256
#define WAVES 8
#define ITERS 8
#define OUT_PER_ITER 32
#define OUT_PER_WAVE (ITERS * OUT_PER_ITER)   // 256
#define OUT_PER_BLOCK (WAVES * OUT_PER_WAVE)  // 2048
#define CHUNKS_PER_ROW (NN / OUT_PER_BLOCK)   // 512
#define HALO_L 3
#define XBUF (OUT_PER_BLOCK + HALO_L + 1)     // 2052
#define TRS 33                                // padded transpose stride
#define EPSW 1e-13f

typedef __attribute__((ext_vector_type(2))) float v2f;
typedef __attribute__((ext_vector_type(8))) float v8f;

// 16x4 A-matrix: rows 0-5 = S (taps 0..3), rows 6-8 = L/6 (taps 0..3), rest 0.
// Tap k=4 is applied as a VALU fixup after the WMMA (rows 4,5: +1; row 8: -1/6).
__device__ const float WENO_COEF[16][4] = {
    { 1.f, -2.f,  1.f,  0.f},
    { 1.f, -4.f,  3.f,  0.f},
    { 0.f,  1.f, -2.f,  1.f},
    { 0.f,  1.f,  0.f, -1.f},
    { 0.f,  0.f,  1.f, -2.f},
    { 0.f,  0.f,  3.f, -4.f},
    { 2.f/6.f, -7.f/6.f, 11.f/6.f, 0.f},
    { 0.f, -1.f/6.f,  5.f/6.f,  2.f/6.f},
    { 0.f,  0.f,  2.f/6.f,  5.f/6.f},
    {0.f,0.f,0.f,0.f},{0.f,0.f,0.f,0.f},{0.f,0.f,0.f,0.f},{0.f,0.f,0.f,0.f},
    {0.f,0.f,0.f,0.f},{0.f,0.f,0.f,0.f},{0.f,0.f,0.f,0.f}
};

#if __has_builtin(__builtin_amdgcn_global_load_async_to_lds_b32)
#define HAVE_ASYNC_LDS 1
typedef __attribute__((address_space(1))) int av1_int;   // global
typedef __attribute__((address_space(3))) int av3_int;   // LDS
#endif

__global__ __launch_bounds__(BLOCK_THREADS)
void weno_z_wmma_kernel(const float* __restrict__ x, float* __restrict__ out) {
    __shared__ float xs[XBUF + 4];
    __shared__ float tr[WAVES][9 * TRS];

    const int tid   = threadIdx.x;
    const int lane  = tid & 31;
    const int wv    = tid >> 5;
    const int blk   = blockIdx.x;
    const int batch = blk / CHUNKS_PER_ROW;
    const int start = (blk % CHUNKS_PER_ROW) * OUT_PER_BLOCK;

    const float* xrow = x + (size_t)batch * NN;

    // ---- Stage 1: fill xs[i] = x[(start - 3 + i) & NMASK], i in [0, XBUF) ----
    for (int i = tid; i < XBUF; i += BLOCK_THREADS) {
        const int n = (start - HALO_L + i) & NMASK;
#ifdef HAVE_ASYNC_LDS
        av1_int* gp = (av1_int*)(uintptr_t)(const void*)(xrow + n);
        av3_int* lp = (av3_int*)(uint32_t)(uintptr_t)(const void*)&xs[i];
        __builtin_amdgcn_global_load_async_to_lds_b32(gp, lp, 0, 0);
#else
        xs[i] = xrow[n];
#endif
    }
#ifdef HAVE_ASYNC_LDS
#if __has_builtin(__builtin_amdgcn_s_wait_asynccnt)
    __builtin_amdgcn_s_wait_asynccnt(0);
#else
    asm volatile("s_wait_asynccnt 0" ::: "memory");
#endif
#endif
    __syncthreads();

    // ---- Constant A operand (per-lane): M = lane&15, K = {kh*2, kh*2+1} ----
    const int m  = lane & 15;
    const int kh = (lane >> 4) & 1;   // 0: K=0,1   1: K=2,3
    const int k0 = 2 * kh;
    v2f A;
    A.x = WENO_COEF[m][k0];
    A.y = WENO_COEF[m][k0 + 1];

    const int col = lane & 15;        // N index inside a 16-wide WMMA tile
    float* trw = tr[wv];
    float* orow = out + (size_t)batch * NN + start;

    for (int it = 0; it < ITERS; ++it) {
        const int out0 = wv * OUT_PER_WAVE + it * OUT_PER_ITER;

#pragma unroll
        for (int jj = 0; jj < 2; ++jj) {
            // B operand: B[k][col] = x[n0 + col + k - 3] = xs[out0 + jj*16 + col + k]
            const int q = out0 + jj * 16 + col;
            v2f Bm;
            Bm.x = xs[q + k0];
            Bm.y = xs[q + k0 + 1];

            v8f C = {};
            C = __builtin_amdgcn_wmma_f32_16x16x4_f32(
                    /*neg_a=*/false, A, /*neg_b=*/false, Bm,
                    /*c_mod=*/(short)0, C, /*reuse_a=*/false, /*reuse_b=*/false);

            // Tap k=4 fixup: value x[n+1] = xs[q + 4]
            const float t4 = xs[q + 4];
            const int cg = jj * 16 + col;
            if (lane < 16) {
                // rows 0..7 live here: rows 4,5 get +t4
                C[4] += t4;
                C[5] += t4;
#pragma unroll
                for (int r = 0; r < 8; ++r) trw[r * TRS + cg] = C[r];
            } else {
                // row 8 (P2) lives in C[0] of the upper half: -t4/6
                trw[8 * TRS + cg] = fmaf(t4, -(1.0f / 6.0f), C[0]);
            }
        }

        // ---- Nonlinear WENO-Z: each lane finishes one element (column=lane) ----
        const float a0 = trw[0 * TRS + lane];
        const float a1 = trw[1 * TRS + lane];
        const float a2 = trw[2 * TRS + lane];
        const float a3 = trw[3 * TRS + lane];
        const float a4 = trw[4 * TRS + lane];
        const float a5 = trw[5 * TRS + lane];
        const float p0 = trw[6 * TRS + lane];
        const float p1 = trw[7 * TRS + lane];
        const float p2 = trw[8 * TRS + lane];

        const float is0 = (13.0f / 12.0f) * a0 * a0 + 0.25f * a1 * a1;
        const float is1 = (13.0f / 12.0f) * a2 * a2 + 0.25f * a3 * a3;
        const float is2 = (13.0f / 12.0f) * a4 * a4 + 0.25f * a5 * a5;
        const float t5  = fabsf(is2 - is0);

        const float w0 = 0.1f * (1.0f + t5 * __builtin_amdgcn_rcpf(is0 + EPSW));
        const float w1 = 0.6f * (1.0f + t5 * __builtin_amdgcn_rcpf(is1 + EPSW));
        const float w2 = 0.3f * (1.0f + t5 * __builtin_amdgcn_rcpf(is2 + EPSW));
        const float inv = __builtin_amdgcn_rcpf(w0 + w1 + w2);

        orow[out0 + lane] = (w0 * p0 + w1 * p1 + w2 * p2) * inv;
    }
}

extern "C" void kernel_launch(void* const* d_in, const int* in_sizes, int n_in,
                              void* d_out, int out_size, void* d_ws, size_t ws_size,
                              hipStream_t stream) {
    (void)n_in; (void)d_ws; (void)ws_size; (void)out_size;
    const float* x = (const float*)d_in[0];
    float* out = (float*)d_out;
    const int total = in_sizes[0];                 // B * N = 16 * 2^20
    const int blocks = total / OUT_PER_BLOCK;      // 8192
    weno_z_wmma_kernel<<<dim3(blocks), dim3(BLOCK_THREADS), 0, stream>>>(x, out);
}